// MFLPN_51032801411333
// MI455X (gfx1250) — compile-verified
//
#include <hip/hip_runtime.h>
#include <hip/hip_bf16.h>
#include <math.h>

typedef __attribute__((ext_vector_type(2))) float v2f;
typedef __attribute__((ext_vector_type(8))) float v8f;

__device__ __forceinline__ v8f wmma_f32(v2f a, v2f b, v8f c) {
  // V_WMMA_F32_16X16X4_F32 : full-precision fp32 matrix op (CDNA5)
  return __builtin_amdgcn_wmma_f32_16x16x4_f32(
      /*neg_a=*/false, a, /*neg_b=*/false, b,
      /*c_mod=*/(short)0, c, /*reuse_a=*/false, /*reuse_b=*/false);
}

// gfx1250 async global->LDS copy (ASYNCcnt path); LDS offset = low 32 bits of
// the generic pointer (aperture truncation rule, ISA 10.2).
__device__ __forceinline__ void async_ld_f32(float* lds_dst, const float* g) {
  unsigned int l = (unsigned int)(uintptr_t)lds_dst;
  asm volatile("global_load_async_to_lds_b32 %0, %1, off" :: "v"(l), "v"(g) : "memory");
}
__device__ __forceinline__ void wait_async0() {
  asm volatile("s_wait_asynccnt 0" ::: "memory");
}

__device__ __forceinline__ void atomicMaxFloat(float* addr, float val) {
  unsigned int* a = (unsigned int*)addr;
  unsigned int old = __hip_atomic_load(a, __ATOMIC_RELAXED, __HIP_MEMORY_SCOPE_AGENT);
  while (__uint_as_float(old) < val) {
    unsigned int assumed = old;
    old = atomicCAS(a, assumed, __float_as_uint(val));
    if (old == assumed) break;
  }
}

// ---------------------------------------------------------------- utility
__global__ void fill_f32(float* __restrict__ p, float v, int n) {
  int i = blockIdx.x * blockDim.x + threadIdx.x;
  if (i < n) p[i] = v;
}

// ---------------------------------------------------------------- pre + CNN
__global__ void pre_cnn_kernel(const float* __restrict__ x, const float* __restrict__ rd,
    const float* p1w, const float* p1b, const float* p2w, const float* p2b,
    const float* p3w, const float* p3b,
    const float* c1w, const float* c1b, const float* c2w, const float* c2b,
    const float* c3w, const float* c3b, const float* c4w, const float* c4b,
    const float* c5w, const float* c5b, const float* c6w, const float* c6b,
    float* __restrict__ xp, int n) {
  int i = blockIdx.x * blockDim.x + threadIdx.x;
  if (i >= n) return;
  float out[16];
  float x0 = x[i * 6 + 0], x1v = x[i * 6 + 1];
  for (int j = 0; j < 4; ++j) out[j]     = fmaxf(x0 * p1w[j] + p1b[j], 0.f);
  for (int j = 0; j < 4; ++j) out[4 + j] = fmaxf(x1v * p2w[j] + p2b[j], 0.f);
  for (int j = 0; j < 4; ++j) {
    float v = p3b[j];
    for (int k = 0; k < 4; ++k) v += x[i * 6 + 2 + k] * p3w[k * 4 + j];
    out[8 + j] = fmaxf(v, 0.f);
  }
  // circular pad H by 5 : pad[42][5], node is fastest dim in rd -> coalesced
  float pad[42][5];
  for (int h = 0; h < 42; ++h) {
    int hh = (h + 27) & 31;  // (h-5) mod 32
    for (int w = 0; w < 5; ++w) pad[h][w] = rd[(hh * 5 + w) * n + i];
  }
  float c1[4][40][3];
  for (int o = 0; o < 4; ++o)
    for (int h = 0; h < 40; ++h)
      for (int w = 0; w < 3; ++w) {
        float v = c1b[o];
        for (int dh = 0; dh < 3; ++dh)
          for (int dw = 0; dw < 3; ++dw)
            v += pad[h + dh][w + dw] * c1w[(o * 3 + dh) * 3 + dw];
        c1[o][h][w] = fmaxf(v, 0.f);
      }
  float c2[4][38][3];
  for (int o = 0; o < 4; ++o)
    for (int h = 0; h < 38; ++h)
      for (int w = 0; w < 3; ++w) {
        float v = c2b[o];
        for (int ci = 0; ci < 4; ++ci)
          for (int dh = 0; dh < 3; ++dh)
            v += c1[ci][h + dh][w] * c2w[(o * 4 + ci) * 3 + dh];
        c2[o][h][w] = fmaxf(v, 0.f);
      }
  float c3[4][18][3];
  for (int o = 0; o < 4; ++o)
    for (int h = 0; h < 18; ++h)
      for (int w = 0; w < 3; ++w) {
        float v = c3b[o];
        for (int ci = 0; ci < 4; ++ci)
          for (int dh = 0; dh < 4; ++dh)
            v += c2[ci][2 * h + dh][w] * c3w[(o * 4 + ci) * 4 + dh];
        c3[o][h][w] = fmaxf(v, 0.f);
      }
  float c4[4][8][3];
  for (int o = 0; o < 4; ++o)
    for (int h = 0; h < 8; ++h)
      for (int w = 0; w < 3; ++w) {
        float v = c4b[o];
        for (int ci = 0; ci < 4; ++ci)
          for (int dh = 0; dh < 4; ++dh)
            v += c3[ci][2 * h + dh][w] * c4w[(o * 4 + ci) * 4 + dh];
        c4[o][h][w] = fmaxf(v, 0.f);
      }
  float c5[4][6][3];
  for (int o = 0; o < 4; ++o)
    for (int h = 0; h < 6; ++h)
      for (int w = 0; w < 3; ++w) {
        float v = c5b[o];
        for (int ci = 0; ci < 4; ++ci)
          for (int dh = 0; dh < 3; ++dh)
            v += c4[ci][h + dh][w] * c5w[(o * 4 + ci) * 3 + dh];
        c5[o][h][w] = fmaxf(v, 0.f);
      }
  for (int h = 0; h < 4; ++h) {
    float v = c6b[0];
    for (int ci = 0; ci < 4; ++ci)
      for (int dh = 0; dh < 3; ++dh)
        for (int dw = 0; dw < 3; ++dw)
          v += c5[ci][h + dh][dw] * c6w[(ci * 3 + dh) * 3 + dw];
    out[12 + h] = fmaxf(v, 0.f);
  }
  for (int j = 0; j < 16; ++j) xp[i * 16 + j] = out[j];
}

// ---------------------------------------------------------------- edge stats (once)
__global__ void edge_stats_kernel(const int* __restrict__ dst, const float* __restrict__ ea,
                                  float* __restrict__ deg, float* __restrict__ easum, int E_) {
  int e = blockIdx.x * blockDim.x + threadIdx.x;
  if (e >= E_) return;
  int d = dst[e];
  atomicAdd(&deg[d], 1.f);
  atomicAdd(&easum[d * 2 + 0], ea[e * 2 + 0]);
  atomicAdd(&easum[d * 2 + 1], ea[e * 2 + 1]);
}
__global__ void ea_norm_kernel(float* __restrict__ easum, const float* __restrict__ deg, int n) {
  int i = blockIdx.x * blockDim.x + threadIdx.x;
  if (i >= n) return;
  float dg = fmaxf(deg[i], 1.f);
  easum[i * 2 + 0] /= dg;
  easum[i * 2 + 1] /= dg;
}

// ---------------------------------------------------------------- node GEMM: xs = X @ W  (WMMA f32, B in LDS)
__global__ void gat_xs_gemm(const float* __restrict__ X, const float* __restrict__ W,
                            float* __restrict__ XS, int n_rows, int in_c, int out_c) {
  __shared__ float sW[32 * 32];
  int total = in_c * out_c;
  for (int t = threadIdx.x; t < total; t += blockDim.x) async_ld_f32(&sW[t], &W[t]);
  wait_async0();
  __syncthreads();

  int wave = blockIdx.x * (blockDim.x >> 5) + (threadIdx.x >> 5);
  int lane = threadIdx.x & 31;
  int rowBase = wave * 16;
  if (rowBase >= n_rows) return;  // wave-uniform
  int lr = lane & 15;
  int hi = lane >> 4;  // 0: K=k0,k0+1   1: K=k0+2,k0+3
  for (int nt = 0; nt < out_c; nt += 16) {
    v8f acc = {};
    for (int k0 = 0; k0 < in_c; k0 += 4) {
      int ka = k0 + hi * 2;
      int row = rowBase + lr;
      v2f a, b;
      a.x = X[(size_t)row * in_c + ka];
      a.y = X[(size_t)row * in_c + ka + 1];
      b.x = sW[ka * out_c + nt + lr];
      b.y = sW[(ka + 1) * out_c + nt + lr];
      acc = wmma_f32(a, b, acc);
    }
    int col = nt + lr;
#pragma unroll
    for (int r = 0; r < 8; ++r) {
      int row = rowBase + r + hi * 8;
      XS[(size_t)row * out_c + col] = acc[r];
    }
  }
}

// ---------------------------------------------------------------- attention scalars
__global__ void attn_coef_kernel(const float* __restrict__ xs, const float* atts, const float* attd,
                                 float* __restrict__ a_src, float* __restrict__ a_dst, int n, int C) {
  int i = blockIdx.x * blockDim.x + threadIdx.x;
  if (i >= n) return;
  float s = 0.f, d = 0.f;
  for (int c = 0; c < C; ++c) {
    float v = xs[(size_t)i * C + c];
    s += v * atts[c];
    d += v * attd[c];
  }
  a_src[i] = s;
  a_dst[i] = d;
}

// ---------------------------------------------------------------- layer-constant q = lin_edge @ att_edge
__global__ void compute_q_kernel(const float* __restrict__ linE, const float* __restrict__ attE,
                                 float* __restrict__ q, int C) {
  if (blockIdx.x == 0 && threadIdx.x == 0) {
    float q0 = 0.f, q1 = 0.f;
    for (int c = 0; c < C; ++c) { q0 += linE[c] * attE[c]; q1 += linE[C + c] * attE[c]; }
    q[0] = q0; q[1] = q1;
  }
}

// ---------------------------------------------------------------- alpha + segment max (E real + N self loops)
__global__ void alpha_max_kernel(const int* __restrict__ src, const int* __restrict__ dst,
    const float* __restrict__ ea, const float* __restrict__ eamean,
    const float* __restrict__ a_src, const float* __restrict__ a_dst,
    const float* __restrict__ q,
    float* __restrict__ alpha, float* __restrict__ amax, int E_, int n) {
  int t = blockIdx.x * blockDim.x + threadIdx.x;
  if (t >= E_ + n) return;
  int s, d;
  float e0, e1;
  if (t < E_) { s = src[t]; d = dst[t]; e0 = ea[t * 2]; e1 = ea[t * 2 + 1]; }
  else        { s = d = t - E_; e0 = eamean[(size_t)(t - E_) * 2]; e1 = eamean[(size_t)(t - E_) * 2 + 1]; }
  float a = a_src[s] + a_dst[d] + e0 * q[0] + e1 * q[1];
  a = (a > 0.f) ? a : 0.2f * a;  // leaky_relu(0.2)
  alpha[t] = a;
  atomicMaxFloat(&amax[d], a);
}

// ---------------------------------------------------------------- exp + segment sum
__global__ void exp_sum_kernel(const int* __restrict__ dst, const float* __restrict__ alpha,
                               const float* __restrict__ amax, float* __restrict__ ex,
                               float* __restrict__ denom, int E_, int n) {
  int t = blockIdx.x * blockDim.x + threadIdx.x;
  if (t >= E_ + n) return;
  int d = (t < E_) ? dst[t] : t - E_;
  float v = __expf(alpha[t] - amax[d]);
  ex[t] = v;
  atomicAdd(&denom[d], v);
}

// ---------------------------------------------------------------- edge message GEMM (WMMA, B in LDS) + scatter
__device__ __forceinline__ float edge_a_elem(const float* __restrict__ xs, const int* __restrict__ src,
    const float* __restrict__ ea, const float* fcw, const float* fcb, int e, int k, int C) {
  if (k < C) return xs[(size_t)src[e] * C + k];
  int j = k - C;
  return ea[e * 2] * fcw[j] + ea[e * 2 + 1] * fcw[8 + j] + fcb[j];
}

__global__ void gat_edge_msg(const float* __restrict__ xs, const int* __restrict__ src,
    const int* __restrict__ dst, const float* __restrict__ ea,
    const float* fcw, const float* fcb, const float* __restrict__ lenw, const float* lenb,
    const float* __restrict__ ex, const float* __restrict__ denom,
    float* __restrict__ agg, int E_, int C, int K) {
  __shared__ float sW[40 * 32];
  int total = K * C;
  for (int t = threadIdx.x; t < total; t += blockDim.x) async_ld_f32(&sW[t], &lenw[t]);
  wait_async0();
  __syncthreads();

  int wave = blockIdx.x * (blockDim.x >> 5) + (threadIdx.x >> 5);
  int lane = threadIdx.x & 31;
  int eBase = wave * 16;
  if (eBase >= E_) return;  // wave-uniform
  int lr = lane & 15;
  int hi = lane >> 4;

  // per-lane softmax scales + dst indices for the 8 rows this lane stores
  float alv[8];
  int dn[8];
#pragma unroll
  for (int r = 0; r < 8; ++r) {
    int e = eBase + r + hi * 8;
    int d = dst[e];
    dn[r] = d;
    alv[r] = ex[e] / denom[d];
  }

  for (int nt = 0; nt < C; nt += 16) {
    v8f acc = {};
    for (int k0 = 0; k0 < K; k0 += 4) {
      int ka = k0 + hi * 2;
      int e = eBase + lr;
      v2f a, b;
      a.x = edge_a_elem(xs, src, ea, fcw, fcb, e, ka, C);
      a.y = edge_a_elem(xs, src, ea, fcw, fcb, e, ka + 1, C);
      b.x = sW[ka * C + nt + lr];
      b.y = sW[(ka + 1) * C + nt + lr];
      acc = wmma_f32(a, b, acc);
    }
    int col = nt + lr;
    float bcol = lenb[col];
#pragma unroll
    for (int r = 0; r < 8; ++r) {
      float v = fmaxf(acc[r] + bcol, 0.f);   // relu(n_lin)
      atomicAdd(&agg[(size_t)dn[r] * C + col], v * alv[r]);
    }
  }
}

// ---------------------------------------------------------------- finalize: self loop + bias (+relu)
__global__ void finalize_kernel(const float* __restrict__ agg, const float* __restrict__ xs,
    const float* __restrict__ ex, const float* __restrict__ denom, const float* bias,
    float* __restrict__ z, int n, int C, int E_, int doRelu) {
  int t = blockIdx.x * blockDim.x + threadIdx.x;
  if (t >= n * C) return;
  int i = t / C, c = t - i * C;
  float self = (ex[E_ + i] / denom[i]) * xs[(size_t)i * C + c];
  float v = agg[t] + self + bias[c];
  if (doRelu) v = fmaxf(v, 0.f);
  z[t] = v;
}

// ---------------------------------------------------------------- prediction l1 GEMM (WMMA, K=48, B in LDS)
__device__ __forceinline__ float pred_a_elem(const float* __restrict__ z, const int* __restrict__ sp,
    const int* __restrict__ dp, const float* __restrict__ epf, const float* efw, const float* efb,
    int e, int k) {
  if (k < 16) return z[(size_t)sp[e] * 16 + k];
  if (k < 32) return z[(size_t)dp[e] * 16 + (k - 16)];
  int j = k - 32;
  return epf[e * 3] * efw[j] + epf[e * 3 + 1] * efw[16 + j] + epf[e * 3 + 2] * efw[32 + j] + efb[j];
}

__global__ void pred_l1_gemm(const float* __restrict__ z, const int* __restrict__ sp,
    const int* __restrict__ dp, const float* __restrict__ epf,
    const float* efw, const float* efb, const float* __restrict__ l1w, const float* l1b,
    float* __restrict__ h1, int EP_) {
  __shared__ float sW[48 * 16];
  for (int t = threadIdx.x; t < 48 * 16; t += blockDim.x) async_ld_f32(&sW[t], &l1w[t]);
  wait_async0();
  __syncthreads();

  int wave = blockIdx.x * (blockDim.x >> 5) + (threadIdx.x >> 5);
  int lane = threadIdx.x & 31;
  int eBase = wave * 16;
  if (eBase >= EP_) return;
  int lr = lane & 15;
  int hi = lane >> 4;
  v8f acc = {};
  for (int k0 = 0; k0 < 48; k0 += 4) {
    int ka = k0 + hi * 2;
    int e = eBase + lr;
    v2f a, b;
    a.x = pred_a_elem(z, sp, dp, epf, efw, efb, e, ka);
    a.y = pred_a_elem(z, sp, dp, epf, efw, efb, e, ka + 1);
    b.x = sW[ka * 16 + lr];
    b.y = sW[(ka + 1) * 16 + lr];
    acc = wmma_f32(a, b, acc);
  }
  float bb = l1b[lr];
#pragma unroll
  for (int r = 0; r < 8; ++r) {
    int e = eBase + r + hi * 8;
    h1[(size_t)e * 16 + lr] = fmaxf(acc[r] + bb, 0.f);
  }
}

// ---------------------------------------------------------------- prediction tail l2/l3
__global__ void pred_tail_kernel(const float* __restrict__ h1, const float* l2w, const float* l2b,
                                 const float* l3w, const float* l3b, float* __restrict__ out, int EP_) {
  int e = blockIdx.x * blockDim.x + threadIdx.x;
  if (e >= EP_) return;
  float h[16];
  for (int j = 0; j < 16; ++j) h[j] = h1[(size_t)e * 16 + j];
  float g[8];
  for (int j = 0; j < 8; ++j) {
    float v = l2b[j];
    for (int k = 0; k < 16; ++k) v += h[k] * l2w[k * 8 + j];
    g[j] = fmaxf(v, 0.f);
  }
  for (int j = 0; j < 3; ++j) {
    float v = l3b[j];
    for (int k = 0; k < 8; ++k) v += g[k] * l3w[k * 3 + j];
    out[(size_t)e * 3 + j] = v;
  }
}

// ================================================================ host
extern "C" void kernel_launch(void* const* d_in, const int* in_sizes, int n_in,
                              void* d_out, int out_size, void* d_ws, size_t ws_size,
                              hipStream_t stream) {
  const float* X   = (const float*)d_in[0];   // [N,6]
  const float* RD  = (const float*)d_in[1];   // [32,5,N]
  const float* EA  = (const float*)d_in[2];   // [E,2]
  const float* EPF = (const float*)d_in[3];   // [EP,3]
  const int N  = in_sizes[0] / 6;
  const int E  = in_sizes[2] / 2;
  const int EP = in_sizes[3] / 3;

  int pi = 4;
  const float* p1w = (const float*)d_in[pi++]; const float* p1b = (const float*)d_in[pi++];
  const float* p2w = (const float*)d_in[pi++]; const float* p2b = (const float*)d_in[pi++];
  const float* p3w = (const float*)d_in[pi++]; const float* p3b = (const float*)d_in[pi++];
  const float* c1w = (const float*)d_in[pi++]; const float* c1b = (const float*)d_in[pi++];
  const float* c2w = (const float*)d_in[pi++]; const float* c2b = (const float*)d_in[pi++];
  const float* c3w = (const float*)d_in[pi++]; const float* c3b = (const float*)d_in[pi++];
  const float* c4w = (const float*)d_in[pi++]; const float* c4b = (const float*)d_in[pi++];
  const float* c5w = (const float*)d_in[pi++]; const float* c5b = (const float*)d_in[pi++];
  const float* c6w = (const float*)d_in[pi++]; const float* c6b = (const float*)d_in[pi++];
  struct MPp { const float *lin_src, *att_src, *att_dst, *lin_edge, *att_edge,
                           *fc_w, *fc_b, *len_w, *len_b, *bias; };
  MPp mp[3];
  for (int l = 0; l < 3; ++l) {
    mp[l].lin_src  = (const float*)d_in[pi++];
    mp[l].att_src  = (const float*)d_in[pi++];
    mp[l].att_dst  = (const float*)d_in[pi++];
    mp[l].lin_edge = (const float*)d_in[pi++];
    mp[l].att_edge = (const float*)d_in[pi++];
    mp[l].fc_w     = (const float*)d_in[pi++];
    mp[l].fc_b     = (const float*)d_in[pi++];
    mp[l].len_w    = (const float*)d_in[pi++];
    mp[l].len_b    = (const float*)d_in[pi++];
    mp[l].bias     = (const float*)d_in[pi++];
  }
  const float* efw = (const float*)d_in[pi++]; const float* efb = (const float*)d_in[pi++];
  const float* l1w = (const float*)d_in[pi++]; const float* l1b = (const float*)d_in[pi++];
  const float* l2w = (const float*)d_in[pi++]; const float* l2b = (const float*)d_in[pi++];
  const float* l3w = (const float*)d_in[pi++]; const float* l3b = (const float*)d_in[pi++];
  const int* ei  = (const int*)d_in[pi++];   // [2,E]
  const int* eip = (const int*)d_in[pi++];   // [2,EP]
  const int* src = ei;
  const int* dst = ei + E;
  const int* srcp = eip;
  const int* dstp = eip + EP;

  // workspace layout (floats)
  float* w = (float*)d_ws;
  float* xp    = w; w += (size_t)N * 16;
  float* xs    = w; w += (size_t)N * 32;
  float* a_src = w; w += N;
  float* a_dst = w; w += N;
  float* deg   = w; w += N;
  float* eam   = w; w += (size_t)N * 2;
  float* amax  = w; w += N;
  float* denom = w; w += N;
  float* qbuf  = w; w += 2;
  float* alpha = w; w += (size_t)(E + N);
  float* exv   = w; w += (size_t)(E + N);
  float* agg   = w; w += (size_t)N * 32;
  float* zA    = w; w += (size_t)N * 32;
  float* zB    = w; w += (size_t)N * 32;
  float* h1    = w; w += (size_t)EP * 16;

  const int TB = 256;
  auto gr = [](long n, int tb) { return (int)((n + tb - 1) / tb); };

  // once: degree / mean edge-attr
  fill_f32<<<gr(N, TB), TB, 0, stream>>>(deg, 0.f, N);
  fill_f32<<<gr(2 * N, TB), TB, 0, stream>>>(eam, 0.f, 2 * N);
  edge_stats_kernel<<<gr(E, TB), TB, 0, stream>>>(dst, EA, deg, eam, E);
  ea_norm_kernel<<<gr(N, TB), TB, 0, stream>>>(eam, deg, N);

  // node features
  pre_cnn_kernel<<<gr(N, 128), 128, 0, stream>>>(X, RD,
      p1w, p1b, p2w, p2b, p3w, p3b, c1w, c1b, c2w, c2b, c3w, c3b,
      c4w, c4b, c5w, c5b, c6w, c6b, xp, N);

  struct Layer { const float* in; float* out; int in_c; int C; int relu; };
  Layer layers[3] = { {xp, zA, 16, 32, 1}, {zA, zB, 32, 32, 1}, {zB, zA, 32, 16, 0} };

  const float NEG_INF = -3.402823466e38f;
  for (int l = 0; l < 3; ++l) {
    const Layer& L = layers[l];
    const MPp& P = mp[l];
    int C = L.C, K = C + 8;
    fill_f32<<<gr(N, TB), TB, 0, stream>>>(amax, NEG_INF, N);
    fill_f32<<<gr(N, TB), TB, 0, stream>>>(denom, 0.f, N);
    fill_f32<<<gr((long)N * C, TB), TB, 0, stream>>>(agg, 0.f, N * C);
    compute_q_kernel<<<1, 32, 0, stream>>>(P.lin_edge, P.att_edge, qbuf, C);

    int nodeWaves = (N + 15) / 16;                       // N % 16 == 0
    gat_xs_gemm<<<gr(nodeWaves, 8), 256, 0, stream>>>(L.in, P.lin_src, xs, N, L.in_c, C);
    attn_coef_kernel<<<gr(N, TB), TB, 0, stream>>>(xs, P.att_src, P.att_dst, a_src, a_dst, N, C);
    alpha_max_kernel<<<gr((long)E + N, TB), TB, 0, stream>>>(src, dst, EA, eam, a_src, a_dst,
        qbuf, alpha, amax, E, N);
    exp_sum_kernel<<<gr((long)E + N, TB), TB, 0, stream>>>(dst, alpha, amax, exv, denom, E, N);
    int edgeWaves = (E + 15) / 16;                       // E % 16 == 0
    gat_edge_msg<<<gr(edgeWaves, 8), 256, 0, stream>>>(xs, src, dst, EA,
        P.fc_w, P.fc_b, P.len_w, P.len_b, exv, denom, agg, E, C, K);
    finalize_kernel<<<gr((long)N * C, TB), TB, 0, stream>>>(agg, xs, exv, denom, P.bias,
        L.out, N, C, E, L.relu);
  }

  // prediction head (z lives in zA, C=16)
  int predWaves = (EP + 15) / 16;                        // EP % 16 == 0
  pred_l1_gemm<<<gr(predWaves, 8), 256, 0, stream>>>(zA, srcp, dstp, EPF, efw, efb,
      l1w, l1b, h1, EP);
  pred_tail_kernel<<<gr(EP, TB), TB, 0, stream>>>(h1, l2w, l2b, l3w, l3b, (float*)d_out, EP);
}